// GenderClassifier_68607807586810
// MI455X (gfx1250) — compile-verified
//
#include <hip/hip_runtime.h>
#include <hip/hip_bf16.h>

// Problem constants (from reference)
#define IN_DIM   16384      // 4*64*64
#define BATCH    4096
#define OUT_CH   2
#define N_SLICES 32                    // K split for parallelism
#define SLICE_K  (IN_DIM / N_SLICES)   // 512 elements per wave
#define TK       128                   // K-tile staged in LDS
#define TK_PAD   132                   // +4 dwords -> conflict-free ds reads
#define WAVES_PER_BLOCK 2
// per-wave LDS: A tile (16 rows) + W tile (32 rows = 16 samples x 2 ch)
#define LDS_ROWS_PER_WAVE (16 + 32)
#define LDS_F32_PER_WAVE  (LDS_ROWS_PER_WAVE * TK_PAD)

typedef __attribute__((ext_vector_type(2))) float v2f;
typedef __attribute__((ext_vector_type(4))) float v4f;
typedef __attribute__((ext_vector_type(8))) float v8f;

// ---------------------------------------------------------------------------
// Kernel 1: out[b*2+ch] = bias[t[b]][ch]   (also clears the 0xAA poison)
// ---------------------------------------------------------------------------
__global__ __launch_bounds__(256) void moe_bias_init(const int* __restrict__ ts,
                                                     const float* __restrict__ bias,
                                                     float* __restrict__ out) {
    int i = blockIdx.x * 256 + threadIdx.x;
    if (i < BATCH * OUT_CH) {
        int s  = i >> 1;
        int ch = i & 1;
        out[i] = bias[ts[s] * OUT_CH + ch];
    }
}

// ---------------------------------------------------------------------------
// Kernel 2: LDS-staged WMMA-diagonal batched matvec with expert gather.
//
// One wave handles 16 samples x one 512-element K-slice:
//   - stage 16x128 x-tile + 2x16x128 gathered weight tile into LDS with
//     fully coalesced global_load_b128 (512B contiguous per wave-load)
//   - ds_load_b64 the strided WMMA operands (bank-conflict-free, pad=4)
//   - V_WMMA_F32_16X16X4_F32: D[m,m] accumulates dot(x_m, w_{t_m}[ch])
//   - atomically add K-slice partials into bias-initialized out
// ---------------------------------------------------------------------------
__global__ __launch_bounds__(WAVES_PER_BLOCK * 32) void
moe_wmma(const float* __restrict__ x,
         const int*   __restrict__ ts,
         const float* __restrict__ W,
         float*       __restrict__ out) {
    __shared__ float lds[WAVES_PER_BLOCK * LDS_F32_PER_WAVE];

    const int wave = threadIdx.x >> 5;
    const int lane = threadIdx.x & 31;
    float* As = lds + wave * LDS_F32_PER_WAVE;       // 16 rows of x
    float* Ws = As + 16 * TK_PAD;                    // 32 rows of weights

    const int wid   = blockIdx.x * WAVES_PER_BLOCK + wave;   // 0..8191
    const int group = wid >> 5;                      // 0..255 (16-sample group)
    const int slice = wid & 31;                      // 0..31  (K slice)

    const int m    = lane & 15;                      // WMMA row/col index
    const int half = lane >> 4;                      // K-pair select in operand

    // Expert ids for the 16 samples of this group (uniform -> scalar loads)
    int tid16[16];
    #pragma unroll
    for (int r = 0; r < 16; ++r) tid16[r] = ts[group * 16 + r];

    const float* xg = x + (size_t)(group * 16) * IN_DIM + slice * SLICE_K;

    v8f c0 = {};   // channel 0 accumulator (diagonal useful)
    v8f c1 = {};   // channel 1 accumulator

    for (int tile = 0; tile < SLICE_K; tile += TK) {
        // ---- stage x tile: 16 rows, 512B coalesced per row ----
        #pragma unroll
        for (int r = 0; r < 16; ++r) {
            v4f v = *(const v4f*)(xg + (size_t)r * IN_DIM + tile + lane * 4);
            *(v4f*)(As + r * TK_PAD + lane * 4) = v;
        }
        // ---- stage gathered weight tile: 16 samples x 2 channels ----
        #pragma unroll
        for (int r = 0; r < 16; ++r) {
            const float* wr = W + (size_t)tid16[r] * (OUT_CH * IN_DIM)
                                + slice * SLICE_K + tile + lane * 4;
            v4f w0 = *(const v4f*)(wr);
            v4f w1 = *(const v4f*)(wr + IN_DIM);
            *(v4f*)(Ws + r        * TK_PAD + lane * 4) = w0;
            *(v4f*)(Ws + (16 + r) * TK_PAD + lane * 4) = w1;
        }
        // cross-lane LDS RAW: DS ops are in-order per wave; stop compiler
        // from reordering the strided reads above the stores.
        __builtin_amdgcn_wave_barrier();

        // ---- WMMA over the staged tile ----
        #pragma unroll 4
        for (int kk = 0; kk < TK; kk += 4) {
            v2f a  = *(const v2f*)(As + m        * TK_PAD + kk + 2 * half);
            v2f b0 = *(const v2f*)(Ws + m        * TK_PAD + kk + 2 * half);
            v2f b1 = *(const v2f*)(Ws + (16 + m) * TK_PAD + kk + 2 * half);
            c0 = __builtin_amdgcn_wmma_f32_16x16x4_f32(false, a, false, b0,
                                                       (short)0, c0, false, false);
            c1 = __builtin_amdgcn_wmma_f32_16x16x4_f32(false, a, false, b1,
                                                       (short)0, c1, false, false);
        }
        // WAR on LDS before next tile's stores (in-order DS + compiler fence)
        __builtin_amdgcn_wave_barrier();
    }

    // Diagonal extraction (C/D layout: VGPR r: lanes 0-15 -> M=r, N=lane;
    // lanes 16-31 -> M=r+8, N=lane-16).
    const bool active = (lane < 8) || (lane >= 24);
    const int  r      = (lane < 8) ? lane : (lane - 24);   // VGPR index
    const int  mres   = (lane < 8) ? lane : (lane - 16);   // sample-in-group

    float v0 = c0[0], v1 = c1[0];
    #pragma unroll
    for (int i = 1; i < 8; ++i) {
        v0 = (r == i) ? c0[i] : v0;
        v1 = (r == i) ? c1[i] : v1;
    }

    if (active) {
        float* o = out + (size_t)(group * 16 + mres) * OUT_CH;
        atomicAdd(o + 0, v0);
        atomicAdd(o + 1, v1);
    }
}

// ---------------------------------------------------------------------------
// Inputs (setup_inputs order): x (f32), timesteps (int32), W (f32), b (f32)
// Output: 4096*2 f32
// ---------------------------------------------------------------------------
extern "C" void kernel_launch(void* const* d_in, const int* in_sizes, int n_in,
                              void* d_out, int out_size, void* d_ws, size_t ws_size,
                              hipStream_t stream) {
    const float* x   = (const float*)d_in[0];
    const int*   ts  = (const int*)  d_in[1];
    const float* W   = (const float*)d_in[2];
    const float* b   = (const float*)d_in[3];
    float*       out = (float*)d_out;

    (void)in_sizes; (void)n_in; (void)out_size; (void)d_ws; (void)ws_size;

    // 1) out = gathered bias
    moe_bias_init<<<(BATCH * OUT_CH + 255) / 256, 256, 0, stream>>>(ts, b, out);

    // 2) accumulate WMMA-diagonal partial dots (8192 waves, 2 waves/block)
    const int n_waves  = (BATCH / 16) * N_SLICES;          // 8192
    const int n_blocks = n_waves / WAVES_PER_BLOCK;        // 4096
    moe_wmma<<<n_blocks, WAVES_PER_BLOCK * 32, 0, stream>>>(x, ts, W, out);
}